// BasicTJDLayer_28621662060825
// MI455X (gfx1250) — compile-verified
//
#include <hip/hip_runtime.h>
#include <hip/hip_bf16.h>
#include <math.h>

#define RANK   32
#define VOCAB  32000
#define BATCH  16
#define SEQ    1024
#define EPSV   1e-10f
#define SEG_LEN 32
#define NSEG    32   // SEQ / SEG_LEN
#define VCHUNK  256  // vocab rows per reduction block (32000 / 256 = 125)

typedef __attribute__((ext_vector_type(2))) float v2f;
typedef __attribute__((ext_vector_type(8))) float v8f;

// ---------------------------------------------------------------------------
// WMMA helper: one wave computes 16x16 output tile (ti,tj) of C += A(32x32) x B(32x32)
// using V_WMMA_F32_16X16X4_F32 over 8 k-steps of 4.
// Fragment layouts per CDNA5 ISA 7.12.2 (f32):
//   A 16x4 : lanes 0-15 rows, VGPR0={K0 | K2}, VGPR1={K1 | K3} (halves of wave)
//   B 4x16 : cols striped across lanes, K rows split across lane halves
//   C 16x16: 8 VGPRs; lanes 0-15 -> M=vgpr, lanes 16-31 -> M=8+vgpr
// ---------------------------------------------------------------------------
__device__ __forceinline__ v8f wmma_tile_32(const float* __restrict__ A,
                                            const float* __restrict__ B,
                                            int ti, int tj, int lane) {
  v8f c = {0.f, 0.f, 0.f, 0.f, 0.f, 0.f, 0.f, 0.f};
  const int half = lane >> 4;   // 0 or 1
  const int l16  = lane & 15;
  const int arow = ti * 16 + l16;
  const int bcol = tj * 16 + l16;
#pragma unroll
  for (int kt = 0; kt < 8; ++kt) {
    const int k0 = kt * 4 + half * 2;
    v2f a, b;
    a.x = A[arow * 32 + k0];
    a.y = A[arow * 32 + k0 + 1];
    b.x = B[k0 * 32 + bcol];
    b.y = B[(k0 + 1) * 32 + bcol];
    c = __builtin_amdgcn_wmma_f32_16x16x4_f32(false, a, false, b,
                                              (short)0, c, false, false);
  }
  return c;
}

__device__ __forceinline__ void store_tile_32(float* __restrict__ D,
                                              int ti, int tj, int lane, v8f c) {
  const int half = lane >> 4;
  const int l16  = lane & 15;
  const int col  = tj * 16 + l16;
#pragma unroll
  for (int m = 0; m < 8; ++m) {
    const int row = ti * 16 + half * 8 + m;
    D[row * 32 + col] = c[m];
  }
}

// ---------------------------------------------------------------------------
// Kernel 1: zero the M accumulator (1024 floats)
// ---------------------------------------------------------------------------
__global__ void tjd_init_kernel(float* __restrict__ M) {
  const int i = blockIdx.x * 256 + threadIdx.x;
  if (i < RANK * RANK) M[i] = 0.f;
}

// ---------------------------------------------------------------------------
// Kernel 2: M[r][s] = sum_v core[r][v][s]. Streams all 131 MB of core once
// (warms the 192 MB L2 for the gather kernel). grid = (125, 32), block = 256.
// Per-step each block touches a contiguous 1 KB line (tid-linear addressing).
// ---------------------------------------------------------------------------
__global__ void tjd_reduce_m_kernel(const float* __restrict__ core,
                                    float* __restrict__ M) {
  const int r  = blockIdx.y;
  const int s  = threadIdx.x & 31;
  const int vg = threadIdx.x >> 5;     // 0..7
  const int v0 = blockIdx.x * VCHUNK;
  const float* base = core + (size_t)r * VOCAB * RANK;
  float acc = 0.f;
  for (int v = v0 + vg; v < v0 + VCHUNK; v += 8)
    acc += base[(size_t)v * RANK + s];
  __shared__ float red[256];
  red[threadIdx.x] = acc;
  __syncthreads();
  if (threadIdx.x < 32) {
    float sum = 0.f;
#pragma unroll
    for (int g = 0; g < 8; ++g) sum += red[g * 32 + threadIdx.x];
    atomicAdd(&M[r * 32 + threadIdx.x], sum);
  }
}

// ---------------------------------------------------------------------------
// Kernel 3: norm = alpha^T (M^1024) beta via 10 WMMA squarings (1024 = 2^10).
// Single block of 128 threads (4 waves, one 16x16 tile each), LDS ping-pong.
// ---------------------------------------------------------------------------
__global__ void tjd_norm_kernel(const float* __restrict__ M,
                                const float* __restrict__ alpha,
                                const float* __restrict__ beta,
                                float* __restrict__ norm_out) {
  __shared__ float bufA[RANK * RANK];
  __shared__ float bufB[RANK * RANK];
  __shared__ float red[128];
  const int tid  = threadIdx.x;
  const int lane = tid & 31;
  const int wid  = tid >> 5;
  const int ti = wid >> 1, tj = wid & 1;

  for (int i = tid; i < RANK * RANK; i += 128) bufA[i] = M[i];
  __syncthreads();

  float* cur = bufA;
  float* nxt = bufB;
  for (int it = 0; it < 10; ++it) {
    v8f c = wmma_tile_32(cur, cur, ti, tj, lane);
    store_tile_32(nxt, ti, tj, lane, c);
    __syncthreads();
    float* t = cur; cur = nxt; nxt = t;
  }

  float partial = 0.f;
  for (int i = tid; i < RANK * RANK; i += 128) {
    const int r = i >> 5, s = i & 31;
    partial += alpha[r] * cur[i] * beta[s];
  }
  red[tid] = partial;
  __syncthreads();
  for (int off = 64; off > 0; off >>= 1) {
    if (tid < off) red[tid] += red[tid + off];
    __syncthreads();
  }
  if (tid == 0) *norm_out = red[0];
}

// ---------------------------------------------------------------------------
// Kernel 4: segment products. Each block (128 thr, 4 waves) computes
//   P[b][seg] = G_{t0} * G_{t0+1} * ... * G_{t0+31}   (32x32 f32, WMMA)
// Software-pipelined: tile t+1 is fetched from global into registers while
// the WMMA matmul for tile t runs; registers are committed to LDS only at the
// top of the next iteration, so the s_wait_loadcnt lands a full matmul +
// barrier after issue (hides the 32-line strided gather latency).
// grid = (NSEG, BATCH)
// ---------------------------------------------------------------------------
__global__ void tjd_seg_prod_kernel(const float* __restrict__ core,
                                    const int* __restrict__ labels,
                                    float* __restrict__ P) {
  const int b    = blockIdx.y;
  const int seg  = blockIdx.x;
  const int tid  = threadIdx.x;
  const int lane = tid & 31;
  const int wid  = tid >> 5;
  const int ti = wid >> 1, tj = wid & 1;

  __shared__ float G [RANK * RANK];
  __shared__ float Pc[RANK * RANK];
  __shared__ float Pn[RANK * RANK];
  float* cur = Pc;
  float* nxt = Pn;

  const int* lb = labels + b * SEQ + seg * SEG_LEN;

  // Each thread owns 8 elements i = tid + 128*j of the 32x32 tile.
  // For element i: r = i>>5 (fixed per j), s = i&31 = tid&31 (fixed).
  const int s_own = tid & 31;
  const int r0    = tid >> 5;               // r for j: r0 + 4*j
  float regs[8];

  // t = 0: product starts as the first gathered tile (straight to cur)
  {
    const int y0 = lb[0];
#pragma unroll
    for (int j = 0; j < 8; ++j)
      regs[j] = core[((size_t)(r0 + 4 * j) * VOCAB + y0) * RANK + s_own];
#pragma unroll
    for (int j = 0; j < 8; ++j)
      cur[(r0 + 4 * j) * 32 + s_own] = regs[j];
  }
  // issue loads for tile t = 1
  {
    const int y1 = lb[1];
#pragma unroll
    for (int j = 0; j < 8; ++j)
      regs[j] = core[((size_t)(r0 + 4 * j) * VOCAB + y1) * RANK + s_own];
  }
  __syncthreads();

  for (int t = 1; t < SEG_LEN; ++t) {
    // commit tile t (already in registers) to LDS
#pragma unroll
    for (int j = 0; j < 8; ++j)
      G[(r0 + 4 * j) * 32 + s_own] = regs[j];
    __syncthreads();

    // issue loads for tile t+1; latency hidden behind the WMMA matmul below
    if (t + 1 < SEG_LEN) {
      const int yn = lb[t + 1];
#pragma unroll
      for (int j = 0; j < 8; ++j)
        regs[j] = core[((size_t)(r0 + 4 * j) * VOCAB + yn) * RANK + s_own];
    }

    v8f c = wmma_tile_32(cur, G, ti, tj, lane);
    store_tile_32(nxt, ti, tj, lane, c);
    __syncthreads();
    float* tmp = cur; cur = nxt; nxt = tmp;
  }

  float* dst = P + ((size_t)b * NSEG + seg) * (RANK * RANK);
#pragma unroll
  for (int j = 0; j < 8; ++j)
    dst[(r0 + 4 * j) * 32 + s_own] = cur[(r0 + 4 * j) * 32 + s_own];
}

// ---------------------------------------------------------------------------
// Kernel 5: per-batch chain over the 32 segment products (one wave per batch).
//   v = alpha; for seg: v = v @ P[b][seg];  prob_tilde[b] = v . beta
// ---------------------------------------------------------------------------
__global__ void tjd_chain_kernel(const float* __restrict__ P,
                                 const float* __restrict__ alpha,
                                 const float* __restrict__ beta,
                                 float* __restrict__ prob_tilde) {
  const int b = blockIdx.x;
  const int s = threadIdx.x;   // 32 threads
  __shared__ float vsh[RANK];
  float v = alpha[s];
  for (int seg = 0; seg < NSEG; ++seg) {
    vsh[s] = v;
    __syncthreads();
    const float* Pm = P + ((size_t)b * NSEG + seg) * (RANK * RANK);
    float acc = 0.f;
#pragma unroll 8
    for (int r = 0; r < RANK; ++r) acc += vsh[r] * Pm[r * 32 + s];
    __syncthreads();
    v = acc;
  }
  float x = v * beta[s];
#pragma unroll
  for (int off = 16; off > 0; off >>= 1) x += __shfl_xor(x, off, 32);
  if (s == 0) prob_tilde[b] = x;
}

// ---------------------------------------------------------------------------
// Kernel 6: loss = mean(-log(pt+eps) + log(norm+eps)); prob = pt / norm
// d_out = [loss, prob[0..15]]
// ---------------------------------------------------------------------------
__global__ void tjd_finalize_kernel(const float* __restrict__ prob_tilde,
                                    const float* __restrict__ norm_p,
                                    float* __restrict__ out) {
  if (threadIdx.x == 0) {
    const float norm = *norm_p;
    const float logn = logf(norm + EPSV);
    float acc = 0.f;
    for (int b = 0; b < BATCH; ++b) {
      const float p = prob_tilde[b];
      out[1 + b] = p / norm;
      acc += -logf(p + EPSV) + logn;
    }
    out[0] = acc / (float)BATCH;
  }
}

// ---------------------------------------------------------------------------
extern "C" void kernel_launch(void* const* d_in, const int* in_sizes, int n_in,
                              void* d_out, int out_size, void* d_ws, size_t ws_size,
                              hipStream_t stream) {
  const float* alpha  = (const float*)d_in[0];
  const float* beta   = (const float*)d_in[1];
  const float* core   = (const float*)d_in[2];
  const int*   labels = (const int*)d_in[3];
  float* out = (float*)d_out;
  float* ws  = (float*)d_ws;

  // workspace layout (floats)
  float* M     = ws;            // 1024
  float* normp = ws + 1024;     // 1
  float* pt    = ws + 1040;     // 16
  float* P     = ws + 2048;     // BATCH*NSEG*1024 = 524288  (~2 MB total)

  tjd_init_kernel<<<4, 256, 0, stream>>>(M);
  tjd_reduce_m_kernel<<<dim3(VOCAB / VCHUNK, RANK), 256, 0, stream>>>(core, M);
  tjd_norm_kernel<<<1, 128, 0, stream>>>(M, alpha, beta, normp);
  tjd_seg_prod_kernel<<<dim3(NSEG, BATCH), 128, 0, stream>>>(core, labels, P);
  tjd_chain_kernel<<<BATCH, 32, 0, stream>>>(P, alpha, beta, pt);
  tjd_finalize_kernel<<<1, 32, 0, stream>>>(pt, normp, out);
}